// PolyLoss_34711925687064
// MI455X (gfx1250) — compile-verified
//
#include <hip/hip_runtime.h>

#ifndef __has_builtin
#define __has_builtin(x) 0
#endif

#if __has_builtin(__builtin_amdgcn_global_load_async_to_lds_b128)
#define ASYNC_LDS 1
#else
#define ASYNC_LDS 0
#endif

#define GAS __attribute__((address_space(1)))
#define LAS __attribute__((address_space(3)))

typedef int v4i __attribute__((__vector_size__(16)));  // matches builtin param

// Problem constants (reference shape is fixed at 8192x8192 f32).
constexpr int   N    = 8192;
constexpr int   TM   = 64;          // rows per block strip
constexpr int   TN   = 64;          // cols per tile
constexpr int   HALF = N / 2;       // cols per block (grid.x = 2)
constexpr int   NT   = HALF / TN;   // tiles per block = 64
constexpr float NEGF = -1e30f;
constexpr int   SENT = (int)0xFF800000; // -inf bit pattern; very negative as int

#if ASYNC_LDS
__device__ __forceinline__ void wait_async_4() {
#if __has_builtin(__builtin_amdgcn_s_wait_asynccnt)
  __builtin_amdgcn_s_wait_asynccnt(4);
#else
  asm volatile("s_wait_asynccnt 4" ::: "memory");
#endif
}
__device__ __forceinline__ void wait_async_0() {
#if __has_builtin(__builtin_amdgcn_s_wait_asynccnt)
  __builtin_amdgcn_s_wait_asynccnt(0);
#else
  asm volatile("s_wait_asynccnt 0" ::: "memory");
#endif
}
#endif

// Stage one TM x TN f32 tile into LDS. 1024 16-byte chunks, 4 per thread.
// Async path: per-lane GLOBAL_LOAD_ASYNC_TO_LDS_B128 (tracked by ASYNCcnt),
// cpol=1 (TH_NT): stream-once data, keep L2 for the atomic arrays.
__device__ __forceinline__ void fill_tile(const float* __restrict__ S,
                                          float* lbuf, int strip, int tileC,
                                          int tid) {
#pragma unroll
  for (int j = 0; j < 4; ++j) {
    const int c   = tid + 256 * j;   // chunk id 0..1023
    const int r   = c >> 4;          // 16 chunks per 64-float row
    const int k16 = c & 15;
    const float* g = S + (size_t)(strip + r) * N + tileC + k16 * 4;
    float*       l = lbuf + c * 4;
#if ASYNC_LDS
    __builtin_amdgcn_global_load_async_to_lds_b128((GAS v4i*)g, (LAS v4i*)l,
                                                   0, /*cpol: TH_NT*/ 1);
#else
    *(float4*)l = *(const float4*)g;
#endif
  }
}

// One block: rows [strip, strip+64) x cols [colbase, colbase+4096).
// Double-buffered async tile pipeline; row maxes exact per half (shfl reduce),
// col maxes via LDS ds_max_i32 then global_atomic_max_i32 (scores >= 0, so
// float order == int order on the bit patterns).
__global__ __launch_bounds__(256) void poly_rowcol_max(
    const float* __restrict__ S, int* __restrict__ colmax,
    float* __restrict__ rowpart) {
  __shared__ __align__(16) float buf[2][TM * TN];
  __shared__ int colstage[TN];

  const int tx      = threadIdx.x;          // 0..31 (one wave per ty)
  const int ty      = threadIdx.y;          // 0..7
  const int tid     = ty * 32 + tx;
  const int strip   = blockIdx.y * TM;
  const int colbase = blockIdx.x * HALF;

  if (tid < TN) colstage[tid] = SENT;

  float rowAcc[8];
#pragma unroll
  for (int k = 0; k < 8; ++k) rowAcc[k] = NEGF;

  fill_tile(S, buf[0], strip, colbase, tid);  // prologue

  for (int t = 0; t < NT; ++t) {
    if (t + 1 < NT)
      fill_tile(S, buf[(t + 1) & 1], strip, colbase + (t + 1) * TN, tid);
#if ASYNC_LDS
    if (t + 1 < NT) wait_async_4(); else wait_async_0();
#endif
    __syncthreads();  // tile t resident in LDS for all waves

    const float* tb    = buf[t & 1];
    const int    tileC = colbase + t * TN;
    float cx = NEGF, cy = NEGF;

    if (tileC == strip) {  // the single diagonal tile of this strip
#pragma unroll
      for (int k = 0; k < 8; ++k) {
        const int r = ty + 8 * k;
        float2 v = *(const float2*)&tb[r * TN + tx * 2];
        if (tx * 2     == r) v.x = NEGF;  // exclude (i,i)
        if (tx * 2 + 1 == r) v.y = NEGF;
        rowAcc[k] = fmaxf(rowAcc[k], fmaxf(v.x, v.y));
        cx = fmaxf(cx, v.x);
        cy = fmaxf(cy, v.y);
      }
    } else {
#pragma unroll
      for (int k = 0; k < 8; ++k) {
        const int r = ty + 8 * k;
        const float2 v = *(const float2*)&tb[r * TN + tx * 2];
        rowAcc[k] = fmaxf(rowAcc[k], fmaxf(v.x, v.y));
        cx = fmaxf(cx, v.x);
        cy = fmaxf(cy, v.y);
      }
    }

    // Cross-wave column combine in LDS (ds_max_i32; conflict-free banks).
    atomicMax(&colstage[tx * 2],     __float_as_int(cx));
    atomicMax(&colstage[tx * 2 + 1], __float_as_int(cy));
    __syncthreads();

    if (tid < TN) {
      atomicMax(&colmax[tileC + tid], colstage[tid]);  // L2-resident, no HBM
      colstage[tid] = SENT;  // next tile's ds atomics are after next barrier
    }
  }

  // Row maxes: wave32 butterfly reduce over the 32 column lanes.
#pragma unroll
  for (int k = 0; k < 8; ++k) {
    float m = rowAcc[k];
#pragma unroll
    for (int off = 16; off > 0; off >>= 1)
      m = fmaxf(m, __shfl_xor(m, off, 32));
    if (tx == 0) rowpart[blockIdx.x * N + strip + ty + 8 * k] = m;
  }
}

__global__ void poly_init(int* __restrict__ colmax) {
  const int i = blockIdx.x * blockDim.x + threadIdx.x;
  if (i < N) colmax[i] = SENT;
}

// Epilogue: per-row quadratics + deterministic tree reduction.
__global__ __launch_bounds__(256) void poly_finalize(
    const float* __restrict__ S, const int* __restrict__ colmax,
    const float* __restrict__ rowpart, float* __restrict__ out) {
  __shared__ double sacc[256];
  const int tid = threadIdx.x;
  double acc = 0.0;
  for (int i = tid; i < N; i += 256) {
    const float d    = S[(size_t)i * (N + 1)];
    const float rmax = fmaxf(rowpart[i], rowpart[N + i]);
    const float cmax = __int_as_float(colmax[i]);
    if (d < 1.0f - 1e-5f) {  // pos_valid
      const float pl = fmaxf(0.2f * d * d - 0.7f * d + 0.5f, 0.0f);
      if (rmax > d)  // has_neg (row direction); neg_max == rmax
        acc += (double)(pl + fmaxf(0.9f * rmax * rmax - 0.4f * rmax + 0.03f, 0.0f));
      if (cmax > d)  // has_neg (col direction)
        acc += (double)(pl + fmaxf(0.9f * cmax * cmax - 0.4f * cmax + 0.03f, 0.0f));
    }
  }
  sacc[tid] = acc;
  __syncthreads();
  for (int s = 128; s > 0; s >>= 1) {
    if (tid < s) sacc[tid] += sacc[tid + s];
    __syncthreads();
  }
  if (tid == 0) out[0] = (float)(sacc[0] / (double)N);
}

extern "C" void kernel_launch(void* const* d_in, const int* in_sizes, int n_in,
                              void* d_out, int out_size, void* d_ws,
                              size_t ws_size, hipStream_t stream) {
  (void)in_sizes; (void)n_in; (void)out_size; (void)ws_size;
  const float* S = (const float*)d_in[0];
  // Workspace layout (96 KB total): rowpart[2*N] floats, then colmax[N] ints.
  float* rowpart = (float*)d_ws;
  int*   colmax  = (int*)d_ws + 2 * N;

  poly_init<<<dim3((N + 255) / 256), dim3(256), 0, stream>>>(colmax);
  poly_rowcol_max<<<dim3(2, N / TM), dim3(32, 8), 0, stream>>>(S, colmax,
                                                               rowpart);
  poly_finalize<<<dim3(1), dim3(256), 0, stream>>>(S, colmax, rowpart,
                                                   (float*)d_out);
}